// Net_62964220559570
// MI455X (gfx1250) — compile-verified
//
#include <hip/hip_runtime.h>
#include <hip/hip_bf16.h>
#include <stdint.h>

// ---------------------------------------------------------------------------
// Fused Canny-style edge map, MI455X (gfx1250).
// Bandwidth-bound: ~133 MB min traffic -> ~5.7us at 23.3 TB/s. Single fused
// pass; raw channel tiles staged global->LDS with CDNA5 async DMA
// (global_load_async_to_lds_b128 / ASYNCcnt / s_wait_asynccnt); vectorized
// non-temporal B128 output stores.
// ---------------------------------------------------------------------------

#define TW 64              // output tile width  (x)
#define TH 16              // output tile height (y)
#define GW (TW + 4)        // gray tile width  (halo +-2)  = 68
#define GH (TH + 4)        // gray tile height (halo +-2)  = 20
#define RW (TW + 8)        // raw staging width (16B-aligned start) = 72
#define RV (RW / 4)        // vec4 transfers per raw row = 18
#define MW (TW + 2)        // mag tile width  (halo +-1) = 66
#define MH (TH + 2)        // mag tile height (halo +-1) = 18
#define NTHR 256           // 8 wave32 waves

typedef float v4f __attribute__((ext_vector_type(4)));

__global__ __launch_bounds__(NTHR)
void canny_fused_kernel(const float* __restrict__ img,      // (3,H,W)
                        const float* __restrict__ sobel_h,  // (3,3)
                        const float* __restrict__ sobel_v,  // (3,3)
                        const float* __restrict__ dir_w,    // (8,3,3)
                        float* __restrict__ out,            // (H,W)
                        int H, int W) {
    __shared__ __align__(16) float raw[3][GH][RW];  // async-DMA staging, 17.3 KB
    __shared__ float gray[GH][GW];                  // 5.4 KB
    __shared__ float mag[MH][MW];                   // 4.8 KB
    __shared__ float wS[18];                        // sobel_h | sobel_v
    __shared__ float wD[72];                        // 8 directional 3x3 filters

    const int tid = threadIdx.x;
    const int bx  = blockIdx.x * TW;
    const int by  = blockIdx.y * TH;
    const int HW  = H * W;

    // --- filter weights -> LDS (tiny, broadcast) ---------------------------
    if (tid < 9)        wS[tid]      = sobel_h[tid];
    else if (tid < 18)  wS[tid]      = sobel_v[tid - 9];
    else if (tid < 90)  wD[tid - 18] = dir_w[tid - 18];

    // --- async stage 3 raw channel tiles with halo into LDS ----------------
    // rows  by-2 .. by+TH+1   (clamped; gray writer zeroes true OOB by coord)
    // cols  bx-4 .. bx+TW+3   (bx % 64 == 0 -> 16B-aligned global addresses)
    {
        const int NV = 3 * GH * RV;                 // 1080 vec4 transfers
        for (int i = tid; i < NV; i += NTHR) {
            int v  = i % RV;
            int r  = (i / RV) % GH;
            int ch = i / (RV * GH);
            int gy = by + r - 2;       gy = min(max(gy, 0), H - 1);
            int gc = bx + 4 * v - 4;   gc = min(max(gc, 0), W - 4);
            const float* gsrc = img + (size_t)ch * HW + (size_t)gy * W + gc;
            uint32_t lds_off = (uint32_t)(uintptr_t)&raw[ch][r][4 * v];
            asm volatile("global_load_async_to_lds_b128 %0, %1, off"
                         :: "v"(lds_off), "v"((uint64_t)(uintptr_t)gsrc)
                         : "memory");
        }
        asm volatile("s_wait_asynccnt 0" ::: "memory");   // ASYNCcnt == 0
    }
    __syncthreads();

    // --- grayscale tile (zero outside image: SAME-conv zero padding) -------
    for (int i = tid; i < GH * GW; i += NTHR) {
        int r = i / GW, c = i % GW;
        int gy = by + r - 2, gx = bx + c - 2;
        float g = 0.0f;
        if (gy >= 0 && gy < H && gx >= 0 && gx < W) {
            int cc = c + 2;  // raw covers cols bx-4.., gray covers bx-2..
            g = (raw[0][r][cc] + raw[1][r][cc] + raw[2][r][cc]) * (1.0f / 3.0f);
        }
        gray[r][c] = g;
    }
    __syncthreads();

    // --- sobel weights into registers (uniform) ----------------------------
    float wsx[9], wsy[9];
#pragma unroll
    for (int k = 0; k < 9; ++k) { wsx[k] = wS[k]; wsy[k] = wS[9 + k]; }

    // --- gradient magnitude tile with +-1 halo (zero outside image) --------
    for (int i = tid; i < MH * MW; i += NTHR) {
        int r = i / MW, c = i % MW;
        int gy = by + r - 1, gx = bx + c - 1;
        float m = 0.0f;
        if (gy >= 0 && gy < H && gx >= 0 && gx < W) {
            float sx = 0.0f, sy = 0.0f;
#pragma unroll
            for (int dy = 0; dy < 3; ++dy)
#pragma unroll
                for (int dx = 0; dx < 3; ++dx) {
                    float g = gray[r + dy][c + dx];
                    sx = fmaf(g, wsx[dy * 3 + dx], sx);
                    sy = fmaf(g, wsy[dy * 3 + dx], sy);
                }
            m = sqrtf(sx * sx + sy * sy);
        }
        mag[r][c] = m;
    }
    __syncthreads();

    // --- orientation, NMS select, threshold, vectorized NT store -----------
    // Each thread owns 4 adjacent pixels of one row: shared 3x6 LDS patches,
    // one 16B non-temporal store (output is write-once, keep L2 for halos).
    {
        const float RAD2DEG = 180.0f / 3.14159f;   // module constant, not pi
        const int r  = tid >> 4;                   // 0..15
        const int c0 = (tid & 15) << 2;            // 0,4,...,60

        float gbuf[3][6];                          // gray rows r+1..r+3
#pragma unroll
        for (int dy = 0; dy < 3; ++dy)
#pragma unroll
            for (int dx = 0; dx < 6; ++dx)
                gbuf[dy][dx] = gray[r + 1 + dy][c0 + 1 + dx];

        float mbuf[3][6];                          // mag rows r..r+2
#pragma unroll
        for (int dy = 0; dy < 3; ++dy)
#pragma unroll
            for (int dx = 0; dx < 6; ++dx)
                mbuf[dy][dx] = mag[r + dy][c0 + dx];

        v4f res;
#pragma unroll
        for (int j = 0; j < 4; ++j) {
            // gradients at output pixel (by+r, bx+c0+j)
            float sx = 0.0f, sy = 0.0f;
#pragma unroll
            for (int dy = 0; dy < 3; ++dy)
#pragma unroll
                for (int dx = 0; dx < 3; ++dx) {
                    float g = gbuf[dy][j + dx];
                    sx = fmaf(g, wsx[dy * 3 + dx], sx);
                    sy = fmaf(g, wsy[dy * 3 + dx], sy);
                }

            // octant index: round-half-even like jnp.round; [0,8] -> &7
            float ang = fmaf(atan2f(sy, sx), RAD2DEG, 180.0f);
            float o   = rintf(ang * (1.0f / 45.0f));
            int ip  = ((int)o) & 7;
            int ing = (ip + 4) & 7;

            float pos = 0.0f, neg = 0.0f;
#pragma unroll
            for (int dy = 0; dy < 3; ++dy)
#pragma unroll
                for (int dx = 0; dx < 3; ++dx) {
                    float m = mbuf[dy][j + dx];
                    int k = dy * 3 + dx;
                    pos = fmaf(m, wD[ip  * 9 + k], pos);
                    neg = fmaf(m, wD[ing * 9 + k], neg);
                }

            float mc = mbuf[1][j + 1];
            res[j] = (fminf(pos, neg) > 0.0f && mc >= 0.2f) ? 1.0f : 0.0f;
        }

        // 16B-aligned (W%4==0, bx%64==0, c0%4==0) non-temporal store
        v4f* dst = (v4f*)(out + (size_t)(by + r) * W + (bx + c0));
        __builtin_nontemporal_store(res, dst);
    }
}

extern "C" void kernel_launch(void* const* d_in, const int* in_sizes, int n_in,
                              void* d_out, int out_size, void* d_ws, size_t ws_size,
                              hipStream_t stream) {
    const float* img     = (const float*)d_in[0];   // (1,3,H,W) f32
    const float* sobel_h = (const float*)d_in[1];   // (1,1,3,3) f32
    const float* sobel_v = (const float*)d_in[2];   // (1,1,3,3) f32
    const float* dir_w   = (const float*)d_in[3];   // (8,1,3,3) f32
    float* out = (float*)d_out;                     // (1,1,H,W) f32

    const int W = 3840;
    const int H = in_sizes[0] / (3 * W);            // 2160

    dim3 grid(W / TW, H / TH);                      // 60 x 135, exact cover
    canny_fused_kernel<<<grid, NTHR, 0, stream>>>(img, sobel_h, sobel_v,
                                                  dir_w, out, H, W);
}